// hGLM_32770600469057
// MI455X (gfx1250) — compile-verified
//
#include <hip/hip_runtime.h>
#include <hip/hip_bf16.h>

// ---------------------------------------------------------------------------
// hGLM forward for MI455X (gfx1250, wave32).
//
// Pipeline:
//   K0: zero the 50 padding rows at each end of S (workspace).
//   K1: S[t+50, c] = sum_{n : n % 92 == c} X[t, n]   via V_WMMA_F32_16X16X4_F32
//       (one-hot B operand -> fp32-exact segmented reduction on the matrix unit;
//        one WMMA per 4 K-elements, 23-periodic accumulator pattern unrolled).
//   K2: temporal conv (101 taps, 4 groups) over S + sigmoid subunit tree.
// ---------------------------------------------------------------------------

#define T_STEPS 30000
#define N_SYNS  1500
#define M_SUB   23
#define G_GRP   4
#define B_LEN   101
#define PADR    50
#define SROWS   (T_STEPS + 2 * PADR)   // 30100
#define SCOLS   96                      // 92 classes padded to 96 (6 x 16)
#define KSTEPS  (N_SYNS / 4)            // 375 = 16*23 + 7

typedef float v2f __attribute__((ext_vector_type(2)));
typedef float v8f __attribute__((ext_vector_type(8)));

// ---------------------------------------------------------------- kernel 0 --
__global__ void hglm_zero_pad(float* __restrict__ S) {
  int i = blockIdx.x * blockDim.x + threadIdx.x;
  if (i < 2 * PADR * SCOLS) {
    int j = (i < PADR * SCOLS) ? i
                               : ((T_STEPS + PADR) * SCOLS + (i - PADR * SCOLS));
    S[j] = 0.0f;
  }
}

// ---------------------------------------------------------------- kernel 1 --
// One wave per 16-timestep tile. D(16x16) tiles: rows = time, cols = class.
// For k-step ks (covering synapses n=4ks..4ks+3): classes are 4p..4p+3 with
// p = ks % 23, all inside column-tile jt = p>>2, and the one-hot B operand
// depends only on q = p & 3  ->  4 precomputed register B matrices.
__global__ __launch_bounds__(32) void hglm_reduce92_wmma(
    const float* __restrict__ X, float* __restrict__ S) {
  const int lane = threadIdx.x & 31;
  const int half = lane >> 4;   // K-half selector for A/B operands
  const int nidx = lane & 15;   // M index (A) / N index (B, D)
  const int t0   = blockIdx.x * 16;

  // B_q[k, j] = (j == 4q + k) ? 1 : 0   (4x16, f32, 2 VGPRs per lane)
  v2f bmat[4];
#pragma unroll
  for (int q = 0; q < 4; ++q) {
    bmat[q].x = (nidx == 4 * q + 2 * half)     ? 1.0f : 0.0f;  // K = 2*half
    bmat[q].y = (nidx == 4 * q + 2 * half + 1) ? 1.0f : 0.0f;  // K = 2*half+1
  }

  v8f acc[6];
#pragma unroll
  for (int j = 0; j < 6; ++j)
#pragma unroll
    for (int e = 0; e < 8; ++e) acc[j][e] = 0.0f;

  // A operand: lane holds row M = nidx of X-tile, K pair (2*half, 2*half+1).
  const float* ap = X + (size_t)(t0 + nidx) * N_SYNS + 2 * half;

#pragma unroll 1
  for (int kb = 0; kb < 368; kb += 23) {   // 16 full periods of 23
#pragma unroll
    for (int p = 0; p < 23; ++p) {
      v2f a = *(const v2f*)(ap + 4 * (kb + p));
      acc[p >> 2] = __builtin_amdgcn_wmma_f32_16x16x4_f32(
          false, a, false, bmat[p & 3], (short)0, acc[p >> 2], false, false);
    }
  }
#pragma unroll
  for (int p = 0; p < 7; ++p) {            // tail: ks = 368..374
    v2f a = *(const v2f*)(ap + 4 * (368 + p));
    acc[p >> 2] = __builtin_amdgcn_wmma_f32_16x16x4_f32(
        false, a, false, bmat[p & 3], (short)0, acc[p >> 2], false, false);
  }

  // D layout: VGPR r -> M = r + 8*half, N = nidx.
#pragma unroll
  for (int jt = 0; jt < 6; ++jt)
#pragma unroll
    for (int r = 0; r < 8; ++r)
      S[(size_t)(t0 + PADR + r + 8 * half) * SCOLS + jt * 16 + nidx] = acc[jt][r];
}

// ---------------------------------------------------------------- kernel 2 --
#define TT      56                 // timesteps per block
#define LROWS   (TT + B_LEN - 1)   // 156 staged rows
#define LSTRIDE 100                // padded row stride (banks: 100 % 64 = 36)

__global__ __launch_bounds__(64) void hglm_conv_tree(
    const float* __restrict__ S,   const float* __restrict__ Vo,
    const float* __restrict__ Tau, const float* __restrict__ Delta,
    const float* __restrict__ W,   const float* __restrict__ C,
    const float* __restrict__ Theta, float* __restrict__ out) {
  __shared__ __align__(16) float sS[LROWS * LSTRIDE];
  __shared__ float swk[G_GRP * B_LEN];
  __shared__ float sExpC[M_SUB];
  __shared__ float sTheta[M_SUB];

  const int tid = threadIdx.x;
  const int t0  = blockIdx.x * TT;

  // Stage S rows [t0, t0+LROWS) into LDS (24 float4 per row).
  for (int i = tid; i < LROWS * 24; i += 64) {
    int r = i / 24, c4 = i % 24;
    size_t grow = (size_t)(t0 + r);
    float4 v = make_float4(0.0f, 0.0f, 0.0f, 0.0f);
    if (grow < (size_t)SROWS) v = *(const float4*)(S + grow * SCOLS + c4 * 4);
    *(float4*)(sS + r * LSTRIDE + c4 * 4) = v;
  }

  // Per-group temporal kernels (404 values), scaled by group weight.
  {
    float edel = expf(Delta[0]);
    for (int j = tid; j < G_GRP * B_LEN; j += 64) {
      int g = j / B_LEN, b = j % B_LEN;
      float tb   = fmaxf((float)b - edel, 0.0f);
      float tau  = expf(Tau[g]);
      float tf   = tb / tau;
      float fast = tf * expf(-tf);
      float kv;
      if (g < G_GRP / 2) {
        float ts   = tb / (tau * 2.8f + 10.4f);
        float slow = ts * expf(-ts) * 0.3f;
        kv = (fast + slow) / 1.3f;
      } else {
        kv = fast;
      }
      swk[j] = kv * W[g];
    }
    for (int j = tid; j < M_SUB; j += 64) {
      sExpC[j]  = expf(C[j]);
      sTheta[j] = Theta[j];
    }
  }
  __syncthreads();

  const int t = t0 + tid;
  if (tid >= TT || t >= T_STEPS) return;

  float sub[M_SUB];
#pragma unroll
  for (int m = 0; m < M_SUB; ++m) sub[m] = 0.0f;

  for (int b = 0; b < B_LEN; ++b) {
    const float* rp = sS + (tid + b) * LSTRIDE;
    float w0 = swk[b];
    float w1 = swk[B_LEN + b];
    float w2 = swk[2 * B_LEN + b];
    float w3 = swk[3 * B_LEN + b];
#pragma unroll
    for (int m = 0; m < M_SUB; ++m) {
      float4 v = *(const float4*)(rp + 4 * m);   // classes 4m..4m+3 (g=0..3)
      sub[m] += w0 * v.x + w1 * v.y + w2 * v.z + w3 * v.w;
    }
  }

  // Subunit tree: leaves 11..22, internal nodes r=10..0 with children 2r+1, 2r+2.
  float o[M_SUB];
#pragma unroll
  for (int m = 11; m < M_SUB; ++m)
    o[m] = sExpC[m] / (1.0f + expf(-(sub[m] - sTheta[m])));
#pragma unroll
  for (int r = 10; r >= 0; --r) {
    float s = sub[r] + o[2 * r + 1] + o[2 * r + 2];
    o[r] = sExpC[r] / (1.0f + expf(-(s - sTheta[r])));
  }
  out[t] = o[0] + Vo[0];
}

// --------------------------------------------------------------- launcher --
extern "C" void kernel_launch(void* const* d_in, const int* in_sizes, int n_in,
                              void* d_out, int out_size, void* d_ws, size_t ws_size,
                              hipStream_t stream) {
  const float* X     = (const float*)d_in[0];
  const float* Vo    = (const float*)d_in[1];
  const float* Tau   = (const float*)d_in[2];
  const float* Delta = (const float*)d_in[3];
  const float* W     = (const float*)d_in[4];
  const float* C     = (const float*)d_in[5];
  const float* Theta = (const float*)d_in[6];
  float* S   = (float*)d_ws;     // SROWS x SCOLS fp32  (~11.6 MB)
  float* out = (float*)d_out;    // T_STEPS fp32

  hglm_zero_pad<<<(2 * PADR * SCOLS + 255) / 256, 256, 0, stream>>>(S);
  hglm_reduce92_wmma<<<T_STEPS / 16, 32, 0, stream>>>(X, S);
  hglm_conv_tree<<<(T_STEPS + TT - 1) / TT, 64, 0, stream>>>(
      S, Vo, Tau, Delta, W, C, Theta, out);
}